// TdSuperGATNet_69063074119745
// MI455X (gfx1250) — compile-verified
//
#include <hip/hip_runtime.h>

#define DIM 128
#define HEADS 8
#define NEG_SLOPE 0.2f

typedef float v2f __attribute__((ext_vector_type(2)));
typedef float v8f __attribute__((ext_vector_type(8)));

// ---------------------------------------------------------------------------
// Wt[n][k] = W[k][n]  (128x128, 64KB — negligible cost, enables b64 B loads)
// ---------------------------------------------------------------------------
__global__ __launch_bounds__(256) void transpose128_k(const float* __restrict__ W,
                                                      float* __restrict__ Wt) {
    const int i = blockIdx.x * blockDim.x + threadIdx.x;   // 0..16383
    if (i >= DIM * DIM) return;
    const int k = i >> 7, n = i & (DIM - 1);
    Wt[(size_t)n * DIM + k] = W[i];
}

// ---------------------------------------------------------------------------
// GEMM: H[nrows x 128] = A[nrows x 128] @ W[128 x 128]   (W given transposed)
// One wave per 32x16 output block (two 16x16 WMMA tiles sharing B),
// 8 waves/block cover all 128 columns of a 32-row panel.
// fp32 WMMA (V_WMMA_F32_16X16X4_F32), K=4 per step, 2 independent acc chains.
// ---------------------------------------------------------------------------
__global__ __launch_bounds__(256) void gemm128_k(const float* __restrict__ A,
                                                 const float* __restrict__ Wt,
                                                 float* __restrict__ H,
                                                 int nrows) {
    const int lane = threadIdx.x & 31;
    const int wave = threadIdx.x >> 5;       // 0..7 -> 16-column tile
    const int m0   = blockIdx.x * 32;
    const int n0   = wave * 16;
    if (m0 >= nrows) return;
    const int mrow = lane & 15;
    const int half = lane >> 4;

    const int ra = m0 + mrow;
    const int rb = m0 + 16 + mrow;
    const float* __restrict__ a0row = A + (size_t)(ra < nrows ? ra : nrows - 1) * DIM;
    const float* __restrict__ a1row = A + (size_t)(rb < nrows ? rb : nrows - 1) * DIM;
    const float* __restrict__ brow  = Wt + (size_t)(n0 + mrow) * DIM;

    v8f c0 = {}, c1 = {};
    // A lane layout: M = lane&15, K = v + 2*(lane>>4)
    // B lane layout: N = lane&15, K = r + 2*(lane>>4)
    #pragma unroll
    for (int k0 = 0; k0 < DIM; k0 += 4) {
        const int k = k0 + 2 * half;
        const v2f b  = *(const v2f*)(brow  + k);
        const v2f a0 = *(const v2f*)(a0row + k);
        const v2f a1 = *(const v2f*)(a1row + k);
        c0 = __builtin_amdgcn_wmma_f32_16x16x4_f32(false, a0, false, b,
                                                   (short)0, c0, false, false);
        c1 = __builtin_amdgcn_wmma_f32_16x16x4_f32(false, a1, false, b,
                                                   (short)0, c1, false, false);
    }

    // C/D layout: element r -> M = r + 8*half, N = mrow
    float* __restrict__ out = H + (size_t)m0 * DIM + n0 + mrow;
    #pragma unroll
    for (int r = 0; r < 8; ++r) {
        const int mr = r + 8 * half;
        if (m0 + mr < nrows)      out[(size_t)mr * DIM]        = c0[r];
        if (m0 + 16 + mr < nrows) out[(size_t)(mr + 16) * DIM] = c1[r];
    }
}

// ---------------------------------------------------------------------------
// Per-(node,head) attention scores: al = <h, a_src>, ad = <h, a_dst>
// ---------------------------------------------------------------------------
__global__ __launch_bounds__(256) void scores_k(const float* __restrict__ H,
                                                const float* __restrict__ a_src,
                                                const float* __restrict__ a_dst,
                                                float* __restrict__ al,
                                                float* __restrict__ ad,
                                                int n) {
    const int idx = blockIdx.x * blockDim.x + threadIdx.x;   // node*8 + head
    if (idx >= n * HEADS) return;
    const int hd   = idx & (HEADS - 1);
    const int node = idx >> 3;
    const float* __restrict__ hp = H + (size_t)node * DIM + hd * 16;
    const float* __restrict__ as = a_src + hd * 16;
    const float* __restrict__ av = a_dst + hd * 16;
    float sl = 0.f, sd = 0.f;
    #pragma unroll
    for (int c = 0; c < 16; ++c) {
        const float v = hp[c];
        sl += v * as[c];
        sd += v * av[c];
    }
    al[idx] = sl;
    ad[idx] = sd;
}

// ---------------------------------------------------------------------------
// Init: m = -inf, z = 0 (N*8), agg = 0 (N*128). Launched with N*128 threads.
// ---------------------------------------------------------------------------
__global__ __launch_bounds__(256) void init_k(float* __restrict__ m,
                                              float* __restrict__ z,
                                              float* __restrict__ agg,
                                              int n) {
    const int i = blockIdx.x * blockDim.x + threadIdx.x;
    if (i < n * HEADS) { m[i] = -__builtin_inff(); z[i] = 0.f; }
    if (i < n * DIM)   agg[i] = 0.f;
}

__device__ __forceinline__ float leaky(float v) {
    return v > 0.f ? v : v * NEG_SLOPE;
}

// Sign-aware float atomic max via int/uint ordering (init must be -inf).
__device__ __forceinline__ void atomicMaxF(float* addr, float val) {
    if (val >= 0.f) atomicMax((int*)addr, __float_as_int(val));
    else            atomicMin((unsigned int*)addr, __float_as_uint(val));
}

__device__ __forceinline__ void edge_sd(const int* __restrict__ src,
                                        const int* __restrict__ dst,
                                        int E, int e, int& s, int& d) {
    if (e < E) { s = src[e]; d = dst[e]; }
    else       { s = d = e - E; }                    // self loop
}

// ---------------------------------------------------------------------------
// Pass 1: segment max over incoming edges
// ---------------------------------------------------------------------------
__global__ __launch_bounds__(256) void edge_max_k(const int* __restrict__ src,
                                                  const int* __restrict__ dst,
                                                  int E, int n,
                                                  const float* __restrict__ al,
                                                  const float* __restrict__ ad,
                                                  float* __restrict__ m) {
    const int e = blockIdx.x * blockDim.x + threadIdx.x;
    if (e >= E + n) return;
    int s, d;
    edge_sd(src, dst, E, e, s, d);
    const float* __restrict__ als = al + (size_t)s * HEADS;
    const float* __restrict__ add = ad + (size_t)d * HEADS;
    float* __restrict__ md = m + (size_t)d * HEADS;
    #pragma unroll
    for (int hd = 0; hd < HEADS; ++hd)
        atomicMaxF(&md[hd], leaky(als[hd] + add[hd]));
}

// ---------------------------------------------------------------------------
// Pass 2: z = segment sum of exp(e - m)
// ---------------------------------------------------------------------------
__global__ __launch_bounds__(256) void edge_sum_k(const int* __restrict__ src,
                                                  const int* __restrict__ dst,
                                                  int E, int n,
                                                  const float* __restrict__ al,
                                                  const float* __restrict__ ad,
                                                  const float* __restrict__ m,
                                                  float* __restrict__ z) {
    const int e = blockIdx.x * blockDim.x + threadIdx.x;
    if (e >= E + n) return;
    int s, d;
    edge_sd(src, dst, E, e, s, d);
    const float* __restrict__ als = al + (size_t)s * HEADS;
    const float* __restrict__ add = ad + (size_t)d * HEADS;
    const float* __restrict__ md  = m  + (size_t)d * HEADS;
    float* __restrict__ zd = z + (size_t)d * HEADS;
    #pragma unroll
    for (int hd = 0; hd < HEADS; ++hd)
        atomicAdd(&zd[hd], __expf(leaky(als[hd] + add[hd]) - md[hd]));
}

// ---------------------------------------------------------------------------
// Pass 3: weighted scatter, one thread per (edge, head), 16 channels each
// ---------------------------------------------------------------------------
__global__ __launch_bounds__(256) void edge_scatter_k(const int* __restrict__ src,
                                                      const int* __restrict__ dst,
                                                      int E, int n,
                                                      const float* __restrict__ al,
                                                      const float* __restrict__ ad,
                                                      const float* __restrict__ m,
                                                      const float* __restrict__ z,
                                                      const float* __restrict__ H,
                                                      float* __restrict__ agg) {
    const long long idx = (long long)blockIdx.x * blockDim.x + threadIdx.x;
    if (idx >= (long long)(E + n) * HEADS) return;
    const int hd = (int)(idx & (HEADS - 1));
    const int e  = (int)(idx >> 3);
    int s, d;
    edge_sd(src, dst, E, e, s, d);
    const float sc    = leaky(al[(size_t)s * HEADS + hd] + ad[(size_t)d * HEADS + hd]);
    const float alpha = __expf(sc - m[(size_t)d * HEADS + hd]) /
                        (z[(size_t)d * HEADS + hd] + 1e-16f);
    const float4* __restrict__ hp = (const float4*)(H + (size_t)s * DIM + hd * 16);
    float* __restrict__ op = agg + (size_t)d * DIM + hd * 16;
    #pragma unroll
    for (int j = 0; j < 4; ++j) {
        const float4 hv = hp[j];
        atomicAdd(op + 4 * j + 0, alpha * hv.x);
        atomicAdd(op + 4 * j + 1, alpha * hv.y);
        atomicAdd(op + 4 * j + 2, alpha * hv.z);
        atomicAdd(op + 4 * j + 3, alpha * hv.w);
    }
}

// ---------------------------------------------------------------------------
// x2 = elu(agg + b1)
// ---------------------------------------------------------------------------
__global__ __launch_bounds__(256) void elu_bias_k(const float* __restrict__ agg,
                                                  const float* __restrict__ b,
                                                  float* __restrict__ x2,
                                                  int n) {
    const int i = blockIdx.x * blockDim.x + threadIdx.x;
    if (i >= n * DIM) return;
    const float v = agg[i] + b[i & (DIM - 1)];
    x2[i] = v > 0.f ? v : (__expf(v) - 1.f);
}

// ---------------------------------------------------------------------------
// out[n][c] = mean over heads of agg[n][hd][c] + b2[c]
// ---------------------------------------------------------------------------
__global__ __launch_bounds__(256) void finalize_k(const float* __restrict__ agg,
                                                  const float* __restrict__ b2,
                                                  float* __restrict__ out,
                                                  int n) {
    const int i = blockIdx.x * blockDim.x + threadIdx.x;
    if (i >= n * 16) return;
    const int c    = i & 15;
    const int node = i >> 4;
    const float* __restrict__ ap = agg + (size_t)node * DIM + c;
    float s = 0.f;
    #pragma unroll
    for (int hd = 0; hd < HEADS; ++hd) s += ap[hd * 16];
    out[i] = s * (1.f / HEADS) + b2[c];
}

// ---------------------------------------------------------------------------
extern "C" void kernel_launch(void* const* d_in, const int* in_sizes, int n_in,
                              void* d_out, int out_size, void* d_ws, size_t ws_size,
                              hipStream_t stream) {
    const float* x   = (const float*)d_in[0];
    const int*   ei  = (const int*)d_in[1];
    const float* W1  = (const float*)d_in[2];
    const float* as1 = (const float*)d_in[3];
    const float* ad1 = (const float*)d_in[4];
    const float* b1  = (const float*)d_in[5];
    const float* W2  = (const float*)d_in[6];
    const float* as2 = (const float*)d_in[7];
    const float* ad2 = (const float*)d_in[8];
    const float* b2  = (const float*)d_in[9];

    const int N = in_sizes[0] / DIM;
    const int E = in_sizes[1] / 2;
    const int* src = ei;
    const int* dst = ei + E;
    const int Etot = E + N;

    float* ws  = (float*)d_ws;
    float* h   = ws;                              // [N,128]
    float* agg = h   + (size_t)N * DIM;           // [N,128]
    float* x2  = agg + (size_t)N * DIM;           // [N,128]
    float* al  = x2  + (size_t)N * DIM;           // [N,8]
    float* adw = al  + (size_t)N * HEADS;         // [N,8]
    float* m   = adw + (size_t)N * HEADS;         // [N,8]
    float* z   = m   + (size_t)N * HEADS;         // [N,8]
    float* Wt  = z   + (size_t)N * HEADS;         // [128,128]

    const int TB = 256;
    const int gT     = (DIM * DIM + TB - 1) / TB;
    const int gGemm  = (N + 31) / 32;
    const int gNH    = (N * HEADS + TB - 1) / TB;
    const int gND    = (N * DIM + TB - 1) / TB;
    const int gE     = (Etot + TB - 1) / TB;
    const int gEH    = (int)(((long long)Etot * HEADS + TB - 1) / TB);
    const int gOut   = (N * 16 + TB - 1) / TB;

    // ---- conv1 (concat heads) ----
    transpose128_k<<<gT, TB, 0, stream>>>(W1, Wt);
    gemm128_k     <<<gGemm, TB, 0, stream>>>(x, Wt, h, N);
    scores_k      <<<gNH,   TB, 0, stream>>>(h, as1, ad1, al, adw, N);
    init_k        <<<gND,   TB, 0, stream>>>(m, z, agg, N);
    edge_max_k    <<<gE,    TB, 0, stream>>>(src, dst, E, N, al, adw, m);
    edge_sum_k    <<<gE,    TB, 0, stream>>>(src, dst, E, N, al, adw, m, z);
    edge_scatter_k<<<gEH,   TB, 0, stream>>>(src, dst, E, N, al, adw, m, z, h, agg);
    elu_bias_k    <<<gND,   TB, 0, stream>>>(agg, b1, x2, N);

    // ---- conv2 (mean over heads) ----
    transpose128_k<<<gT, TB, 0, stream>>>(W2, Wt);
    gemm128_k     <<<gGemm, TB, 0, stream>>>(x2, Wt, h, N);
    scores_k      <<<gNH,   TB, 0, stream>>>(h, as2, ad2, al, adw, N);
    init_k        <<<gND,   TB, 0, stream>>>(m, z, agg, N);
    edge_max_k    <<<gE,    TB, 0, stream>>>(src, dst, E, N, al, adw, m);
    edge_sum_k    <<<gE,    TB, 0, stream>>>(src, dst, E, N, al, adw, m, z);
    edge_scatter_k<<<gEH,   TB, 0, stream>>>(src, dst, E, N, al, adw, m, z, h, agg);
    finalize_k    <<<gOut,  TB, 0, stream>>>(agg, b2, (float*)d_out, N);
}